// RoIAlignCustom_90572270338472
// MI455X (gfx1250) — compile-verified
//
#include <hip/hip_runtime.h>
#include <cstdint>

#define B_  4
#define N_  300
#define C_  256
#define H_  200
#define W_  200
#define S_  7
#define HW_ (H_ * W_)

// Issue 196 scattered 4B gathers for one channel's 14x14 tap matrix as
// 7 async global->LDS instructions (per-lane addresses, no VGPR returns,
// tracked by ASYNCcnt).
__device__ __forceinline__ void stage_async(const float* __restrict__ featc,
                                            const int* ys, const int* xs,
                                            float* dst, int lane)
{
#pragma unroll
    for (int t = 0; t < 7; ++t) {
        int e = t * 32 + lane;
        if (e < 196) {
            int r = e / 14;
            int s = e - r * 14;
            const float* gp = featc + (ys[r] * W_ + xs[s]);
            unsigned long long ga = (unsigned long long)(uintptr_t)gp;
            // low 32 bits of a generic pointer into LDS == LDS byte offset
            unsigned la = (unsigned)(uintptr_t)(dst + e);
            asm volatile("global_load_async_to_lds_b32 %0, %1, off"
                         :: "v"(la), "v"(ga)
                         : "memory");
        }
    }
}

__global__ __launch_bounds__(256) void roi_align_async_kernel(
    const float* __restrict__ features,
    const float* __restrict__ boxes,
    float* __restrict__ out)
{
    __shared__ int   s_ys[14];
    __shared__ int   s_xs[14];
    __shared__ float s_wy[14];   // row weights, validity + box_ok folded in
    __shared__ float s_wx[14];   // col weights, validity folded in
    __shared__ float s_buf[8][2][196];  // per-wave double-buffered 14x14 gather

    const int box_id = blockIdx.x;
    const int b   = box_id / N_;
    const int tid = threadIdx.x;

    if (tid < S_) {
        const float* bx = boxes + (size_t)box_id * 4;
        float cx = bx[0], cy = bx[1], w = bx[2], h = bx[3];
        float x1 = (cx - 0.5f * w) * (float)W_;
        float x2 = (cx + 0.5f * w) * (float)W_;
        float y1 = (cy - 0.5f * h) * (float)H_;
        float y2 = (cy + 0.5f * h) * (float)H_;
        float ok = ((x2 > x1) && (y2 > y1)) ? 1.0f : 0.0f;

        int   j = tid;
        float t = ((float)j + 0.5f) * (1.0f / (float)S_);

        // x axis: grid-sample clamp => ix = clamp(px, 0, W) - 0.5
        float px = x1 + t * (x2 - x1);
        float ix = fminf(fmaxf(px, 0.0f), (float)W_) - 0.5f;
        float xf = floorf(ix);
        int   x0 = (int)xf;
        float wx = ix - xf;
        s_xs[2 * j]     = min(max(x0, 0), W_ - 1);
        s_xs[2 * j + 1] = min(max(x0 + 1, 0), W_ - 1);
        s_wx[2 * j]     = (1.0f - wx) * ((x0 >= 0 && x0 < W_) ? 1.0f : 0.0f);
        s_wx[2 * j + 1] = wx * ((x0 + 1 >= 0 && x0 + 1 < W_) ? 1.0f : 0.0f);

        // y axis (fold degenerate-box mask into row weights)
        float py = y1 + t * (y2 - y1);
        float iy = fminf(fmaxf(py, 0.0f), (float)H_) - 0.5f;
        float yf = floorf(iy);
        int   y0 = (int)yf;
        float wy = iy - yf;
        s_ys[2 * j]     = min(max(y0, 0), H_ - 1);
        s_ys[2 * j + 1] = min(max(y0 + 1, 0), H_ - 1);
        s_wy[2 * j]     = (1.0f - wy) * ((y0 >= 0 && y0 < H_) ? 1.0f : 0.0f) * ok;
        s_wy[2 * j + 1] = wy * ((y0 + 1 >= 0 && y0 + 1 < H_) ? 1.0f : 0.0f) * ok;
    }
    __syncthreads();

    const int wave = tid >> 5;
    const int lane = tid & 31;
    const float* featb = features + (size_t)b * C_ * HW_;
    float* outb = out + (size_t)box_id * C_ * (S_ * S_);

    // Prologue: stage first channel of this wave into buffer 0.
    stage_async(featb + (size_t)(wave * 32) * HW_, s_ys, s_xs,
                s_buf[wave][0], lane);

    for (int k = 0; k < 32; ++k) {
        const int c = wave * 32 + k;

        if (k + 1 < 32) {
            // WAR guard: all prior LDS reads of the buffer we are about to
            // overwrite must have returned before new async writes land.
            asm volatile("s_wait_dscnt 0x0" ::: "memory");
            stage_async(featb + (size_t)(c + 1) * HW_, s_ys, s_xs,
                        s_buf[wave][(k + 1) & 1], lane);
            // 7 ops of stage(k+1) may remain in flight; ensure stage(k)'s
            // 7 ops (issued earlier, retire in order) are complete.
            asm volatile("s_wait_asynccnt 0x7" ::: "memory");
        } else {
            asm volatile("s_wait_asynccnt 0x0" ::: "memory");
        }

        const float* G = s_buf[wave][k & 1];
        float* outc = outb + (size_t)c * (S_ * S_);

#pragma unroll
        for (int t2 = 0; t2 < 2; ++t2) {
            int p = t2 * 32 + lane;
            if (p < S_ * S_) {
                int i = p / S_;
                int j = p - i * S_;
                const float* g0 = G + (2 * i) * 14 + 2 * j;
                const float* g1 = G + (2 * i + 1) * 14 + 2 * j;
                float acc = s_wy[2 * i]     * (s_wx[2 * j] * g0[0] + s_wx[2 * j + 1] * g0[1])
                          + s_wy[2 * i + 1] * (s_wx[2 * j] * g1[0] + s_wx[2 * j + 1] * g1[1]);
                outc[p] = acc;
            }
        }
    }
}

extern "C" void kernel_launch(void* const* d_in, const int* in_sizes, int n_in,
                              void* d_out, int out_size, void* d_ws, size_t ws_size,
                              hipStream_t stream) {
    const float* features = (const float*)d_in[0];  // [4,256,200,200] f32
    const float* boxes    = (const float*)d_in[1];  // [4,300,4] f32
    float* out = (float*)d_out;                     // [4,300,256,7,7] f32

    dim3 grid(B_ * N_);   // one block per box
    dim3 block(256);      // 8 wave32s
    hipLaunchKernelGGL(roi_align_async_kernel, grid, block, 0, stream,
                       features, boxes, out);
}